// SparseAttention_40948218200601
// MI455X (gfx1250) — compile-verified
//
#include <hip/hip_runtime.h>
#include <hip/hip_bf16.h>

// ---------------- problem constants (from reference) ----------------
static constexpr int BB = 8;
static constexpr int TT = 2048;
static constexpr int CC = 256;
static constexpr int MTOK = BB * TT;          // 16384 rows
static constexpr int NSEG = 32;               // prefix-scan segments per batch
static constexpr int SEGT = TT / NSEG;        // 64 timesteps per segment

typedef __attribute__((ext_vector_type(2))) float v2f;
typedef __attribute__((ext_vector_type(8))) float v8f;
typedef __attribute__((ext_vector_type(4))) unsigned int u32x4;
typedef __attribute__((ext_vector_type(8))) int i32x8;
typedef __attribute__((ext_vector_type(4))) int i32x4;

// LDS byte offset of a __shared__ object (generic -> AS(3) -> int)
__device__ __forceinline__ unsigned int lds_offset_of(const void* p) {
    return (unsigned int)(unsigned long long)
        (const __attribute__((address_space(3))) void*)p;
}

// ---------------------------------------------------------------------
// TDM 2D tile load: global[tile_y rows x tile_x elems, row stride] -> LDS,
// with per-row padding (pad codes per ISA: interval 0=2..7=256 dwords,
// amount 0=1..127=128 dwords). data_size = 4 bytes (fp32).
// Descriptor packing per CDNA5 ISA 8.3/8.4 (groups 2/3 zero => 2D).
// ---------------------------------------------------------------------
__device__ __forceinline__ void tdm_load_2d_f32(unsigned int lds_byte,
                                                const float* gptr,
                                                unsigned int tile_x,
                                                unsigned int tile_y,
                                                unsigned long long stride_elems,
                                                unsigned int pad_interval_code,
                                                unsigned int pad_amount_code) {
    unsigned long long ga = (unsigned long long)(uintptr_t)gptr;
    u32x4 g0;
    g0[0] = 1u;                                          // count=1 (user D#)
    g0[1] = lds_byte;                                    // lds_addr
    g0[2] = (unsigned int)(ga & 0xffffffffu);            // global_addr[31:0]
    g0[3] = (unsigned int)((ga >> 32) & 0x1ffffffu)      // global_addr[56:32]
            | (2u << 30);                                // type = 2 ("image")

    unsigned int flags = (2u << 16);                     // data_size = 4B
    flags |= (1u << 20)                                  // pad_enable
           | (pad_interval_code << 22)
           | (pad_amount_code << 25);
    i32x8 g1;
    unsigned int td0 = tile_x;                           // tensor_dim0 (tile in-bounds)
    unsigned int td1 = tile_y;                           // tensor_dim1
    g1[0] = (int)flags;                                  // mask=0, flags
    g1[1] = (int)((td0 & 0xffffu) << 16);                // [15:0] atomic addr, [31:16] td0.lo
    g1[2] = (int)((td0 >> 16) | ((td1 & 0xffffu) << 16));
    g1[3] = (int)((td1 >> 16) | (tile_x << 16));         // [31:16] tile_dim0
    g1[4] = (int)(tile_y & 0xffffu);                     // tile_dim1; tile_dim2=0
    g1[5] = (int)(stride_elems & 0xffffffffu);           // tensor_dim0_stride lo
    g1[6] = (int)((stride_elems >> 32) & 0xffffu);       // stride hi; stride1=0
    g1[7] = 0;
    i32x4 z4 = {0, 0, 0, 0};
    i32x8 z8 = {0, 0, 0, 0, 0, 0, 0, 0};
    __builtin_amdgcn_tensor_load_to_lds(g0, g1, z4, z4, z8, 0);
}

// =====================================================================
// Kernel 1/4: C[M,N] = A[M,K] @ W[K,N] + bias[N]   (fp32 WMMA 16x16x4)
// Block tile: 16 (M) x 128 (N). 8 waves, each owns one 16x16 tile.
// Tiles staged via TDM (double-buffered, wave0 issues, s_wait_tensorcnt)
// with TDM-inserted row padding (A: 32->36, W: 128->136 dwords) so WMMA
// fragment reads are LDS-bank-conflict free.
// =====================================================================
#define TK 32
__global__ __launch_bounds__(256)
void gemm_bias_wmma(const float* __restrict__ A, const float* __restrict__ W,
                    const float* __restrict__ bias, float* __restrict__ Cout,
                    int M, int N, int K) {
    __shared__ float As[2][16 * 36];      // rows padded 32 -> 36 floats
    __shared__ float Ws[2][TK * 136];     // rows padded 128 -> 136 floats

    const int tid  = threadIdx.x;
    const int wave = tid >> 5;
    const int lane = tid & 31;
    const int mrow = lane & 15;
    const int koff = (lane < 16) ? 0 : 2;     // K-split across lane halves (ISA layout)
    const int m0   = blockIdx.y * 16;
    const int n0   = blockIdx.x * 128;
    const bool loader = (tid < 32);           // wave 0 (wave-uniform branch)

    const unsigned int asOff0 = lds_offset_of(&As[0][0]);
    const unsigned int asOff1 = lds_offset_of(&As[1][0]);
    const unsigned int wsOff0 = lds_offset_of(&Ws[0][0]);
    const unsigned int wsOff1 = lds_offset_of(&Ws[1][0]);

    v8f acc = {};

    // prologue: DMA chunk 0 into buffer 0
    if (loader) {
        // A tile 16 x 32, row stride K; pad 32 dwords (code 4) by 4 dwords (code 3)
        tdm_load_2d_f32(asOff0, A + (size_t)m0 * K, TK, 16, (unsigned long long)K, 4, 3);
        // W tile 32 x 128, row stride N; pad 128 dwords (code 6) by 8 dwords (code 7)
        tdm_load_2d_f32(wsOff0, W + n0, 128, TK, (unsigned long long)N, 6, 7);
        __builtin_amdgcn_s_wait_tensorcnt(0);
    }
    __syncthreads();

    int buf = 0;
    for (int k0 = 0; k0 < K; k0 += TK) {
        const bool more = (k0 + TK) < K;
        if (loader && more) {   // prefetch next chunk into the other buffer
            unsigned int ao = buf ? asOff0 : asOff1;
            unsigned int wo = buf ? wsOff0 : wsOff1;
            tdm_load_2d_f32(ao, A + (size_t)m0 * K + (k0 + TK), TK, 16,
                            (unsigned long long)K, 4, 3);
            tdm_load_2d_f32(wo, W + (size_t)(k0 + TK) * N + n0, 128, TK,
                            (unsigned long long)N, 6, 7);
        }

        const float* as = As[buf];
        const float* ws = Ws[buf];
        #pragma unroll
        for (int kk = 0; kk < TK; kk += 4) {
            v2f a, b;
            // A fragment 16x4: lane holds A[mrow][kk+koff .. +1]
            a.x = as[mrow * 36 + kk + koff];
            a.y = as[mrow * 36 + kk + koff + 1];
            // B fragment 4x16: lane holds W[kk+koff .. +1][n0 + wave*16 + mrow]
            b.x = ws[(kk + koff) * 136 + wave * 16 + mrow];
            b.y = ws[(kk + koff + 1) * 136 + wave * 16 + mrow];
            acc = __builtin_amdgcn_wmma_f32_16x16x4_f32(
                false, a, false, b, (short)0, acc, false, false);
        }

        if (loader && more) __builtin_amdgcn_s_wait_tensorcnt(0);
        __syncthreads();
        buf ^= 1;
    }

    // D layout: VGPR r -> row (r or r+8 by lane half), col = lane&15 within tile
    const int col   = n0 + wave * 16 + mrow;
    const float bv  = bias[col];
    const int rbase = m0 + ((lane < 16) ? 0 : 8);
    #pragma unroll
    for (int r = 0; r < 8; ++r)
        Cout[(size_t)(rbase + r) * N + col] = acc[r] + bv;
}

// =====================================================================
// Kernels 2a/2b/2c: exclusive prefix sum over T of v = qkv[b,:,2C:3C]
// =====================================================================
__global__ void vseg_sum(const float* __restrict__ qkv, float* __restrict__ segsum) {
    int b = blockIdx.x / NSEG, seg = blockIdx.x % NSEG, c = threadIdx.x;
    const float* vb = qkv + (size_t)b * TT * 3 * CC + 2 * CC + c;
    float s = 0.f;
    int t0 = seg * SEGT;
    #pragma unroll 8
    for (int t = 0; t < SEGT; ++t) s += vb[(size_t)(t0 + t) * 3 * CC];
    segsum[((size_t)b * NSEG + seg) * CC + c] = s;
}

__global__ void vseg_scan(float* __restrict__ segsum) {
    int b = blockIdx.x, c = threadIdx.x;
    float acc = 0.f;
    for (int seg = 0; seg < NSEG; ++seg) {
        size_t idx = ((size_t)b * NSEG + seg) * CC + c;
        float v = segsum[idx];
        segsum[idx] = acc;
        acc += v;
    }
}

__global__ void vprefix(const float* __restrict__ qkv, const float* __restrict__ segsum,
                        float* __restrict__ P) {
    int b = blockIdx.x / NSEG, seg = blockIdx.x % NSEG, c = threadIdx.x;
    float acc = segsum[((size_t)b * NSEG + seg) * CC + c];
    const float* vb = qkv + (size_t)b * TT * 3 * CC + 2 * CC + c;
    float* pb = P + (size_t)b * TT * CC + c;
    int t0 = seg * SEGT;
    #pragma unroll 4
    for (int t = t0; t < t0 + SEGT; ++t) {
        pb[(size_t)t * CC] = acc;           // exclusive: P[t] = sum_{j<t} v[j]
        acc += vb[(size_t)t * 3 * CC];
    }
}

// =====================================================================
// Kernel 3: banded attention with out-of-band zero-score mass.
// One wave32 per token. Online softmax seeded with the out-of-band term
// (m=0, Z=nout, acc=P[t-window]). Dot products reduced via shfl_xor.
// =====================================================================
__global__ __launch_bounds__(256)
void attn_band(const float* __restrict__ qkv, const float* __restrict__ P,
               const float* __restrict__ gate, const int* __restrict__ window_p,
               float* __restrict__ att_out) {
    const int gwave = blockIdx.x * (blockDim.x >> 5) + (threadIdx.x >> 5);
    const int lane  = threadIdx.x & 31;
    const int b = gwave / TT;
    const int t = gwave % TT;
    const int win = *window_p;

    const float* qrow  = qkv + (size_t)(b * TT + t) * 3 * CC;          // q in [0,C)
    const float* kbase = qkv + (size_t)b * TT * 3 * CC + CC;           // k in [C,2C)
    const float* vbase = qkv + (size_t)b * TT * 3 * CC + 2 * CC;       // v in [2C,3C)

    float qreg[CC / 32];
    #pragma unroll
    for (int i = 0; i < CC / 32; ++i) qreg[i] = qrow[lane + 32 * i];

    int jstart = t - win; if (jstart < 0) jstart = 0;
    const int nout = jstart;                 // #causal entries with score 0

    float m, Z, acc[CC / 32];
    if (nout > 0) {
        m = 0.f; Z = (float)nout;
        const float* prow = P + (size_t)(b * TT + jstart) * CC;        // sum_{j<jstart} v
        #pragma unroll
        for (int i = 0; i < CC / 32; ++i) acc[i] = prow[lane + 32 * i];
    } else {
        m = -__builtin_inff(); Z = 0.f;
        #pragma unroll
        for (int i = 0; i < CC / 32; ++i) acc[i] = 0.f;
    }

    const float inv_sqrt_c = 0.0625f;        // 1/sqrt(256)
    for (int j = jstart; j <= t; ++j) {
        const float* krow = kbase + (size_t)j * 3 * CC;
        float part = 0.f;
        #pragma unroll
        for (int i = 0; i < CC / 32; ++i) part += qreg[i] * krow[lane + 32 * i];
        #pragma unroll
        for (int off = 16; off > 0; off >>= 1) part += __shfl_xor(part, off, 32);
        const float s  = part * inv_sqrt_c;
        const float mn = fmaxf(m, s);
        const float sc = __expf(m - mn);     // 0 when m == -inf
        const float w  = __expf(s - mn);
        Z = Z * sc + w;
        const float* vrow = vbase + (size_t)j * 3 * CC;
        #pragma unroll
        for (int i = 0; i < CC / 32; ++i)
            acc[i] = acc[i] * sc + w * vrow[lane + 32 * i];
        m = mn;
    }

    const float invZ = 1.0f / Z;
    float* orow = att_out + (size_t)(b * TT + t) * CC;
    #pragma unroll
    for (int i = 0; i < CC / 32; ++i)
        orow[lane + 32 * i] = gate[lane + 32 * i] * acc[i] * invZ;
}

// =====================================================================
extern "C" void kernel_launch(void* const* d_in, const int* in_sizes, int n_in,
                              void* d_out, int out_size, void* d_ws, size_t ws_size,
                              hipStream_t stream) {
    const float* x     = (const float*)d_in[0];
    const float* Wqkv  = (const float*)d_in[1];
    const float* bqkv  = (const float*)d_in[2];
    const float* Wproj = (const float*)d_in[3];
    const float* bproj = (const float*)d_in[4];
    const float* gate  = (const float*)d_in[5];
    const int*   winp  = (const int*)d_in[6];

    // workspace layout
    char* ws = (char*)d_ws;
    float* qkv    = (float*)ws;                                   // 48 MB
    float* P      = (float*)(ws + (size_t)MTOK * 3 * CC * 4);     // 16 MB
    float* att    = (float*)(ws + (size_t)MTOK * 3 * CC * 4
                                + (size_t)MTOK * CC * 4);         // 16 MB
    float* segsum = (float*)(ws + (size_t)MTOK * 3 * CC * 4
                                + 2 * (size_t)MTOK * CC * 4);     // 256 KB

    // 1) qkv = x @ Wqkv + bqkv    [16384 x 768]
    {
        dim3 grid((3 * CC) / 128, MTOK / 16);
        gemm_bias_wmma<<<grid, 256, 0, stream>>>(x, Wqkv, bqkv, qkv, MTOK, 3 * CC, CC);
    }
    // 2) exclusive prefix sums of v over T
    vseg_sum <<<BB * NSEG, CC, 0, stream>>>(qkv, segsum);
    vseg_scan<<<BB,        CC, 0, stream>>>(segsum);
    vprefix  <<<BB * NSEG, CC, 0, stream>>>(qkv, segsum, P);
    // 3) banded attention (one wave per token) -> att (gate applied)
    attn_band<<<MTOK / 8, 256, 0, stream>>>(qkv, P, gate, winp, att);
    // 4) out = att @ Wproj + bproj   [16384 x 256]
    {
        dim3 grid(CC / 128, MTOK / 16);
        gemm_bias_wmma<<<grid, 256, 0, stream>>>(att, Wproj, bproj, (float*)d_out,
                                                 MTOK, CC, CC);
    }
}